// SingleGateQuantumCircuit_2929167696521
// MI455X (gfx1250) — compile-verified
//
#include <hip/hip_runtime.h>
#include <hip/hip_bf16.h>

typedef float v2f __attribute__((ext_vector_type(2)));
typedef float v8f __attribute__((ext_vector_type(8)));

#define BTOT   32768
#define DIN    256
#define HID    256
#define NGEN   255
#define ROWS   16          // batch rows per workgroup
#define XS     260         // stride for x/h/enc rows (260%64==4 -> conflict-free col reads)
#define PS     20          // matrix plane row stride (20*dL===2 mod 64 unsolvable -> conflict-free)
#define PLANE  (16*PS)     // 320 floats per 16x16 plane
#define MREG   (4*PLANE)   // Mre,Mim,Tre,Tim per matrix = 1280 floats

#define EOFF    0                   // enc: 16 x XS
#define XOFF    (ROWS*XS)           // 4160 (aliased by arena later)
#define HOFF    (2*ROWS*XS)         // 8320 (aliased by arena later)
#define AOFF    (ROWS*XS)           // arena: 8 matrices * MREG = 10240 floats [4160,14400)
#define PSIRE   (AOFF + 8*MREG)     // 14400
#define PSIIM   (PSIRE + 256)       // 14656
#define HSHOFF  (PSIIM + 256)       // 14912
#define LDSF    (HSHOFF + 192)      // 15104 floats = 60.4 KB

// spread 4 bits to even bit positions: bit p -> bit 2p
__device__ __forceinline__ int spread4(int x) {
    return (x & 1) | ((x & 2) << 1) | ((x & 4) << 2) | ((x & 8) << 3);
}

// Complex 16x16 matmul on WMMA: A,B are row-major (stride PS) LDS planes.
// Returns C in WMMA C-layout accumulators (cR=real, cI=imag). 16 v_wmma ops.
__device__ __forceinline__ void cmatmul(const float* __restrict__ Ar, const float* __restrict__ Ai,
                                        const float* __restrict__ Br, const float* __restrict__ Bi,
                                        int n16, int ksel, v8f& cR, v8f& cI)
{
    v8f a1 = {}, a2 = {}, a3 = {};
    #pragma unroll
    for (int k0 = 0; k0 < 16; k0 += 4) {
        const int k = k0 + ksel;
        v2f ar; ar.x = Ar[n16 * PS + k];     ar.y = Ar[n16 * PS + k + 1];
        v2f ai; ai.x = Ai[n16 * PS + k];     ai.y = Ai[n16 * PS + k + 1];
        v2f br; br.x = Br[k * PS + n16];     br.y = Br[(k + 1) * PS + n16];
        v2f bi; bi.x = Bi[k * PS + n16];     bi.y = Bi[(k + 1) * PS + n16];
        a1 = __builtin_amdgcn_wmma_f32_16x16x4_f32(false, ar, false, br, (short)0, a1, false, false);
        a2 = __builtin_amdgcn_wmma_f32_16x16x4_f32(false, ai, false, bi, (short)0, a2, false, false);
        a3 = __builtin_amdgcn_wmma_f32_16x16x4_f32(false, ar, false, bi, (short)0, a3, false, false);
        a3 = __builtin_amdgcn_wmma_f32_16x16x4_f32(false, ai, false, br, (short)0, a3, false, false);
    }
    cR = a1 - a2;   // f32 WMMA has no A/B negation (NEG = CNeg only) -> subtract accumulators
    cI = a3;
}

__global__ __launch_bounds__(256)
void qnn_fused(const float* __restrict__ x,  const float* __restrict__ W1,
               const float* __restrict__ b1, const float* __restrict__ W2,
               const float* __restrict__ b2, const float* __restrict__ A_all,
               const float* __restrict__ B_all, const float* __restrict__ D_all,
               float* __restrict__ out)
{
    __shared__ float lds[LDSF];
    const int tid  = threadIdx.x;
    const int row0 = blockIdx.x * ROWS;

    // ---------------- Phase 0: stage x block (16 x 256) into LDS ----------------
    for (int idx = tid; idx < ROWS * DIN; idx += 256) {
        int r = idx >> 8, c = idx & 255;
        lds[XOFF + r * XS + c] = x[(size_t)(row0 + r) * DIN + c];
    }
    __syncthreads();

    const int wave  = tid >> 5, lane = tid & 31;
    const int n16   = lane & 15;
    const int ksel  = (lane < 16) ? 0 : 2;   // K sub-select for f32 16x16x4 WMMA
    const int rbase = (lane < 16) ? 0 : 8;   // C/D tile: VGPR r holds M=r / M=r+8

    // ---------------- Phase 1: h = silu(x@W1 + b1), WMMA f32 16x16x4 ----------------
    for (int t = 0; t < 2; ++t) {
        const int nt  = wave * 2 + t;
        const int col = nt * 16 + n16;
        v8f acc = {};
        #pragma unroll 4
        for (int k0 = 0; k0 < DIN; k0 += 4) {
            const int k = k0 + ksel;
            v2f a; a.x = lds[XOFF + n16 * XS + k];
                   a.y = lds[XOFF + n16 * XS + k + 1];
            v2f b; b.x = W1[(size_t)k * HID + col];
                   b.y = W1[(size_t)(k + 1) * HID + col];
            acc = __builtin_amdgcn_wmma_f32_16x16x4_f32(false, a, false, b,
                                                        (short)0, acc, false, false);
        }
        const float bb = b1[col];
        #pragma unroll
        for (int r = 0; r < 8; ++r) {
            float z = acc[r] + bb;
            lds[HOFF + (rbase + r) * XS + col] = z / (1.0f + expf(-z));
        }
    }
    __syncthreads();

    // ---------------- Phase 2: enc = h@W2 + b2 (255 cols, col 255 zeroed) ----------------
    for (int t = 0; t < 2; ++t) {
        const int nt  = wave * 2 + t;
        const int col = nt * 16 + n16;
        const bool valid = (col < NGEN);
        v8f acc = {};
        #pragma unroll 4
        for (int k0 = 0; k0 < HID; k0 += 4) {
            const int k = k0 + ksel;
            v2f a; a.x = lds[HOFF + n16 * XS + k];
                   a.y = lds[HOFF + n16 * XS + k + 1];
            v2f b; b.x = valid ? W2[(size_t)k * NGEN + col] : 0.0f;
                   b.y = valid ? W2[(size_t)(k + 1) * NGEN + col] : 0.0f;
            acc = __builtin_amdgcn_wmma_f32_16x16x4_f32(false, a, false, b,
                                                        (short)0, acc, false, false);
        }
        const float bb = valid ? b2[col] : 0.0f;
        #pragma unroll
        for (int r = 0; r < 8; ++r) {
            float v = valid ? (acc[r] + bb) : 0.0f;
            lds[EOFF + (rbase + r) * XS + col] = v;
        }
    }
    __syncthreads();   // enc complete; x/h dead -> arena may alias them

    // ===== expm stage: 2 passes x 8 waves; each wave owns one matrix (all wave-local) =====
    float* const Mre = &lds[AOFF + wave * MREG];
    float* const Mim = Mre + PLANE;
    float* const Tre = Mim + PLANE;
    float* const Tim = Tre + PLANE;

    for (int pass = 0; pass < 2; ++pass) {
        const int m = pass * 8 + wave;               // local batch row / matrix id
        const float* encrow = &lds[EOFF + m * XS];

        // --- per-matrix ||G||_F and scaling: ||G||_F^2 = 16 * sum(c^2) ---
        float ss = 0.0f;
        #pragma unroll
        for (int c = 0; c < 8; ++c) {
            float v = encrow[lane * 8 + c];
            ss += v * v;
        }
        #pragma unroll
        for (int d = 16; d >= 1; d >>= 1) ss += __shfl_xor(ss, d, 32);
        const float normF = 4.0f * sqrtf(ss);
        int sshift = 0; float thr = 1.0f;
        while (normF > thr && sshift < 40) { thr *= 2.0f; ++sshift; }
        const float inv = ldexpf(1.0f, -sshift);

        // --- build M = (i * sum_m c_m P_m) * 2^-s directly into LDS planes ---
        // P_m[i, i^f] = (-i)^popc(ymask) * (-1)^popc(i & (ymask|zmask));
        // string index: bit(2p) = (f^s)_p, bit(2p+1) = s_p
        for (int e = lane; e < 256; e += 32) {
            const int i = e >> 4, j = e & 15;
            const int f = i ^ j;
            float ar = 0.0f, ai = 0.0f;
            const int s0 = (f == 0) ? 1 : 0;
            for (int s4 = s0; s4 < 16; ++s4) {
                const int ymask = s4 & f;
                const int zmask = s4 & ~f & 15;
                const int mm = spread4(f ^ s4) | (spread4(s4) << 1);
                float c = encrow[mm - 1];
                if (__popc(i & (ymask | zmask)) & 1) c = -c;
                switch (__popc(ymask) & 3) {      // multiply by (-i)^nY
                    case 0: ar += c; break;
                    case 1: ai -= c; break;
                    case 2: ar -= c; break;
                    default: ai += c; break;
                }
            }
            Mre[i * PS + j] = -ai * inv;   // G = i * (ar + i*ai)
            Mim[i * PS + j] =  ar * inv;
        }
        // wave-local from here on: LDS ops within a wave are in program order

        // --- Taylor: S = I + M + sum_{k=2..12} M^k/k!  (S stays in WMMA C-layout regs) ---
        v8f Sre, Sim;
        #pragma unroll
        for (int r = 0; r < 8; ++r) {
            const int row = rbase + r;
            Sre[r] = Mre[row * PS + n16] + ((row == n16) ? 1.0f : 0.0f);
            Sim[r] = Mim[row * PS + n16];
        }
        const float* Ar = Mre; const float* Ai = Mim;
        for (int k = 2; k <= 12; ++k) {
            v8f nr, ni;
            cmatmul(Ar, Ai, Mre, Mim, n16, ksel, nr, ni);
            const float rk = 1.0f / (float)k;
            nr = nr * rk; ni = ni * rk;
            Sre += nr; Sim += ni;
            #pragma unroll
            for (int r = 0; r < 8; ++r) {
                const int row = rbase + r;
                Tre[row * PS + n16] = nr[r];
                Tim[row * PS + n16] = ni[r];
            }
            Ar = Tre; Ai = Tim;
        }

        // --- write S over M planes (M dead), then s squarings: S <- S*S ---
        #pragma unroll
        for (int r = 0; r < 8; ++r) {
            const int row = rbase + r;
            Mre[row * PS + n16] = Sre[r];
            Mim[row * PS + n16] = Sim[r];
        }
        for (int q = 0; q < sshift; ++q) {
            v8f nr, ni;
            cmatmul(Mre, Mim, Mre, Mim, n16, ksel, nr, ni);
            #pragma unroll
            for (int r = 0; r < 8; ++r) {
                const int row = rbase + r;
                Mre[row * PS + n16] = nr[r];
                Mim[row * PS + n16] = ni[r];
            }
        }

        // --- psi = U[:,0] ---
        if (lane < 16) {
            lds[PSIRE + m * 16 + lane] = Mre[lane * PS];
            lds[PSIIM + m * 16 + lane] = Mim[lane * PS];
        }
    }

    // ---------------- Observables ----------------
    if (tid < 6) {                      // build H for wire-group w = tid
        const int w = tid;
        const int TR[6] = {1, 2, 2, 3, 3, 3};
        const int TC[6] = {0, 0, 1, 0, 1, 2};
        float hre[16], him[16];
        #pragma unroll
        for (int e = 0; e < 16; ++e) { hre[e] = 0.0f; him[e] = 0.0f; }
        #pragma unroll
        for (int k = 0; k < 6; ++k) {
            const int rr = TR[k], cc = TC[k];
            const float a = A_all[w * 6 + k], b = B_all[w * 6 + k];
            hre[rr * 4 + cc] = a;  him[rr * 4 + cc] = b;
            hre[cc * 4 + rr] = a;  him[cc * 4 + rr] = -b;
        }
        #pragma unroll
        for (int d = 0; d < 3; ++d) hre[d * 4 + d] = 2.0f * D_all[w * 4 + d + 1];
        #pragma unroll
        for (int e = 0; e < 16; ++e) {
            lds[HSHOFF + w * 32 + e]      = hre[e];
            lds[HSHOFF + w * 32 + 16 + e] = him[e];
        }
    }
    __syncthreads();

    if (tid < 96) {
        const int rr = tid / 6, w = tid % 6;
        const int P1[6] = {3, 3, 3, 2, 2, 1};   // bit position of qubit q1 (= 3 - q1)
        const int P2[6] = {2, 1, 0, 1, 0, 0};   // bit position of qubit q2
        const int p1 = P1[w], p2 = P2[w];
        const int m1 = 1 << p1, m2 = 1 << p2;
        const float* pr = &lds[PSIRE + rr * 16];
        const float* pi = &lds[PSIIM + rr * 16];
        const float* Hr = &lds[HSHOFF + w * 32];
        const float* Hi = Hr + 16;
        float acc = 0.0f;
        for (int ii = 0; ii < 16; ++ii) {
            const int a_i = 2 * ((ii >> p1) & 1) + ((ii >> p2) & 1);
            const float xr = pr[ii], xi = pi[ii];
            #pragma unroll
            for (int d = 0; d < 4; ++d) {
                const int jj = ii ^ ((d & 1) ? m1 : 0) ^ ((d & 2) ? m2 : 0);
                const int a_j = 2 * ((jj >> p1) & 1) + ((jj >> p2) & 1);
                const float hr = Hr[a_i * 4 + a_j], hi = Hi[a_i * 4 + a_j];
                const float yr = pr[jj], yi = pi[jj];
                acc += xr * (hr * yr - hi * yi) + xi * (hr * yi + hi * yr);
            }
        }
        out[(size_t)(row0 + rr) * 6 + w] = acc;
    }
}

extern "C" void kernel_launch(void* const* d_in, const int* in_sizes, int n_in,
                              void* d_out, int out_size, void* d_ws, size_t ws_size,
                              hipStream_t stream) {
    const float* x     = (const float*)d_in[0];
    const float* W1    = (const float*)d_in[1];
    const float* b1    = (const float*)d_in[2];
    const float* W2    = (const float*)d_in[3];
    const float* b2    = (const float*)d_in[4];
    const float* A_all = (const float*)d_in[5];
    const float* B_all = (const float*)d_in[6];
    const float* D_all = (const float*)d_in[7];
    float* out = (float*)d_out;
    dim3 grid(BTOT / ROWS), block(256);
    hipLaunchKernelGGL(qnn_fused, grid, block, 0, stream,
                       x, W1, b1, W2, b2, A_all, B_all, D_all, out);
}